// MAGNN_nc_mb_AC_67723044323754
// MI455X (gfx1250) — compile-verified
//
#include <hip/hip_runtime.h>
#include <hip/hip_bf16.h>

// ---------------- types / helpers ----------------
typedef __attribute__((ext_vector_type(2))) float v2f;
typedef __attribute__((ext_vector_type(8))) float v8f;

__device__ inline v8f wmma4(v2f a, v2f b, v8f c) {
    // V_WMMA_F32_16X16X4_F32 : D = A(16x4) * B(4x16) + C(16x16)
    return __builtin_amdgcn_wmma_f32_16x16x4_f32(false, a, false, b, (short)0, c,
                                                 false, false);
}

__device__ inline v8f vzero8() {
    v8f v;
    for (int i = 0; i < 8; i++) v[i] = 0.0f;
    return v;
}

__device__ inline float eluf(float x) { return x > 0.0f ? x : expm1f(x); }

#define N_TOT   8192
#define N0_     4096
#define N1_     4096
#define D0_     128
#define D1_     1024
#define HID_    256
#define EMB_    64
#define AVD_    128
#define NKEEP_  3072
#define NDROP_  1024
#define TAU_    0.8f
#define NEG_BIG (-9.0e15f)

// ---------------- generic WMMA GEMM (one wave = one 16x16 tile) ----------------
// C = op(A(MxK,row-major) @ B(KxN,row-major) + bias), optional batch via z.
// mode: 0 = store, 1 = store elu, 2 = C += elu
__global__ void k_gemm16(const float* __restrict__ A, const float* __restrict__ B,
                         const float* __restrict__ bias, float* __restrict__ C,
                         int M, int N, int K, int mode,
                         long sA, long sB, long sC) {
    int bz = blockIdx.z;
    A += (long)bz * sA; B += (long)bz * sB; C += (long)bz * sC;
    int i0 = blockIdx.y * 16, j0 = blockIdx.x * 16;
    int l = threadIdx.x;
    int ln = l & 15, grp = l >> 4;
    v8f acc = vzero8();
    const float* Ar = A + (long)(i0 + ln) * K;
    for (int kk = 0; kk < K; kk += 4) {
        int k0 = kk + 2 * grp;
        v2f a; a.x = Ar[k0]; a.y = Ar[k0 + 1];
        v2f b; b.x = B[(long)k0 * N + j0 + ln]; b.y = B[(long)(k0 + 1) * N + j0 + ln];
        acc = wmma4(a, b, acc);
    }
    int c = j0 + ln;
    float bv = bias ? bias[c] : 0.0f;
    for (int v = 0; v < 8; v++) {
        int r = i0 + v + 8 * grp;
        float x = acc[v] + bv;
        if (mode >= 1) x = eluf(x);
        long o = (long)r * N + c;
        if (mode == 2) C[o] += x; else C[o] = x;
    }
}

// ---------------- column softmax-std over feat1 ----------------
__global__ void k_colstats(const float* __restrict__ feat1, float* __restrict__ stds) {
    __shared__ float red[256];
    int j = blockIdx.x, tid = threadIdx.x;
    float mx = -3.0e38f;
    for (int i = tid; i < N1_; i += 256) mx = fmaxf(mx, feat1[(long)i * D1_ + j]);
    red[tid] = mx; __syncthreads();
    for (int s = 128; s; s >>= 1) { if (tid < s) red[tid] = fmaxf(red[tid], red[tid + s]); __syncthreads(); }
    mx = red[0]; __syncthreads();
    float s1 = 0.0f, s2 = 0.0f;
    for (int i = tid; i < N1_; i += 256) {
        float e = expf(feat1[(long)i * D1_ + j] - mx);
        s1 += e; s2 += e * e;
    }
    red[tid] = s1; __syncthreads();
    for (int s = 128; s; s >>= 1) { if (tid < s) red[tid] += red[tid + s]; __syncthreads(); }
    s1 = red[0]; __syncthreads();
    red[tid] = s2; __syncthreads();
    for (int s = 128; s; s >>= 1) { if (tid < s) red[tid] += red[tid + s]; __syncthreads(); }
    s2 = red[0];
    if (tid == 0) {
        float ss = s2 / (s1 * s1);                     // sum of softmax^2
        float var = (ss - 1.0f / (float)N1_) / (float)(N1_ - 1);
        stds[j] = sqrtf(fmaxf(var, 0.0f));
    }
}

// rank each column by descending std (ties: lower index first); top-256 kept
__global__ void k_rank(const float* __restrict__ stds, int* __restrict__ colpos,
                       int* __restrict__ istop) {
    int j = blockIdx.x * 256 + threadIdx.x;
    if (j >= D1_) return;
    float sj = stds[j];
    int r = 0;
    for (int k = 0; k < D1_; k++) {
        float sk = stds[k];
        r += (sk > sj) || (sk == sj && k < j);
    }
    int top = (r < HID_) ? 1 : 0;
    istop[j] = top;
    if (top) colpos[j] = r;
}
__global__ void k_delpos(const int* __restrict__ istop, int* __restrict__ colpos) {
    int j = blockIdx.x * 256 + threadIdx.x;
    if (j >= D1_ || istop[j]) return;
    int c = 0;
    for (int k = 0; k < j; k++) c += (istop[k] == 0);
    colpos[j] = c;
}

// scatter feat1 columns into featA (topk order) / featB (del order)
__global__ void k_gather(const float* __restrict__ feat1, const int* __restrict__ istop,
                         const int* __restrict__ colpos, float* __restrict__ featA,
                         float* __restrict__ featB) {
    long idx = (long)blockIdx.x * 256 + threadIdx.x;
    if (idx >= (long)N1_ * D1_) return;
    int i = (int)(idx >> 10), j = (int)(idx & (D1_ - 1));
    float v = feat1[idx];
    if (istop[j]) featA[(long)i * HID_ + colpos[j]] = v;
    else          featB[(long)i * (D1_ - HID_) + colpos[j]] = v;
}

// featK[p] = t0[keep[p]]
__global__ void k_gatherK(const float* __restrict__ t0, const int* __restrict__ keep,
                          float* __restrict__ featK) {
    long idx = (long)blockIdx.x * 256 + threadIdx.x;
    if (idx >= (long)NKEEP_ * HID_) return;
    int p = (int)(idx >> 8), f = (int)(idx & 255);
    featK[idx] = t0[(long)keep[p] * HID_ + f];
}

// row L2 norms of a (M x 256) matrix, one wave per row
__global__ void k_rownorm(const float* __restrict__ Z, float* __restrict__ nrm) {
    int r = blockIdx.x, l = threadIdx.x;
    float s = 0.0f;
    for (int k = l; k < HID_; k += 32) { float x = Z[(long)r * HID_ + k]; s += x * x; }
    for (int d = 16; d >= 1; d >>= 1) s += __shfl_xor(s, d, 32);
    if (l == 0) nrm[r] = sqrtf(s);
}

// ---------------- contrast: sum of exp(cos-sim / tau), WMMA tiles ----------------
__global__ void k_simsum(const float* __restrict__ Z1, const float* __restrict__ Z2,
                         const float* __restrict__ n1, const float* __restrict__ n2,
                         float* __restrict__ S) {
    int i0 = blockIdx.y * 16, j0 = blockIdx.x * 16;
    int l = threadIdx.x, ln = l & 15, grp = l >> 4;
    v8f acc = vzero8();
    const float* z1r = Z1 + (long)(i0 + ln) * HID_;
    const float* z2r = Z2 + (long)(j0 + ln) * HID_;
    for (int kk = 0; kk < HID_; kk += 4) {
        int k0 = kk + 2 * grp;
        v2f a; a.x = z1r[k0]; a.y = z1r[k0 + 1];
        v2f b; b.x = z2r[k0]; b.y = z2r[k0 + 1];   // B = Z2^T
        acc = wmma4(a, b, acc);
    }
    float nj = n2[j0 + ln];
    float s = 0.0f;
    for (int v = 0; v < 8; v++) {
        int i = i0 + v + 8 * grp;
        s += expf(acc[v] / (n1[i] * nj * TAU_));
    }
    for (int d = 16; d >= 1; d >>= 1) s += __shfl_xor(s, d, 32);
    if (l == 0) atomicAdd(S, s);
}

// ---------------- attention call 1 (drop rows only) + loss_ac ----------------
__global__ void k_recon_loss(const float* __restrict__ g, const float* __restrict__ hAll,
                             const float* __restrict__ adj, const int* __restrict__ keep,
                             const int* __restrict__ drop, const float* __restrict__ featK,
                             const float* __restrict__ t0, float* __restrict__ loss_acc) {
    __shared__ float erow[NKEEP_];
    __shared__ float gs[AVD_];
    __shared__ float red[256];
    int tid = threadIdx.x;
    int nd = drop[blockIdx.x];
    float acc = 0.0f;
    for (int h = 0; h < 4; h++) {
        if (tid < AVD_) gs[tid] = g[((long)h * N_TOT + nd) * AVD_ + tid];
        __syncthreads();
        for (int m = tid; m < NKEEP_; m += 256) {
            int src = keep[m];
            const float* hr = hAll + ((long)h * N_TOT + src) * AVD_;
            float dot = 0.0f;
            for (int k = 0; k < AVD_; k++) dot += gs[k] * hr[k];
            float x = eluf(dot);
            erow[m] = (adj[(long)nd * N_TOT + src] > 0.0f) ? x : NEG_BIG;
        }
        __syncthreads();
        float mx = -3.0e38f;
        for (int m = tid; m < NKEEP_; m += 256) mx = fmaxf(mx, erow[m]);
        red[tid] = mx; __syncthreads();
        for (int s = 128; s; s >>= 1) { if (tid < s) red[tid] = fmaxf(red[tid], red[tid + s]); __syncthreads(); }
        mx = red[0]; __syncthreads();
        float sm = 0.0f;
        for (int m = tid; m < NKEEP_; m += 256) sm += expf(erow[m] - mx);
        red[tid] = sm; __syncthreads();
        for (int s = 128; s; s >>= 1) { if (tid < s) red[tid] += red[tid + s]; __syncthreads(); }
        float inv = 1.0f / red[0]; __syncthreads();
        for (int m = tid; m < NKEEP_; m += 256) erow[m] = expf(erow[m] - mx) * inv;
        __syncthreads();
        float local = 0.0f;                    // f = tid (HID_ == blockDim)
        for (int m = 0; m < NKEEP_; m++) local += erow[m] * featK[(long)m * HID_ + tid];
        acc += local;
        __syncthreads();
    }
    float recon = acc * 0.25f;
    float diff = t0[(long)nd * HID_ + tid] - recon;
    red[tid] = diff * diff; __syncthreads();
    for (int s = 128; s; s >>= 1) { if (tid < s) red[tid] += red[tid + s]; __syncthreads(); }
    if (tid == 0) atomicAdd(loss_acc, red[0]);
}

// ---------------- attention call 2: pass A (online row max/sum) ----------------
__global__ void k_att2_stats(const float* __restrict__ g, const float* __restrict__ hAll,
                             const float* __restrict__ adj, float* __restrict__ rowmax,
                             float* __restrict__ rowsum) {
    int h = blockIdx.y, n0 = blockIdx.x * 16;
    int l = threadIdx.x, ln = l & 15, grp = l >> 4;
    const float* gh = g + (long)h * N_TOT * AVD_;
    const float* sh = hAll + (long)h * N_TOT * AVD_;
    v2f afr[32];
    const float* ga = gh + (long)(N0_ + n0 + ln) * AVD_;
    for (int kk = 0; kk < 32; kk++) { afr[kk].x = ga[kk * 4 + 2 * grp]; afr[kk].y = ga[kk * 4 + 2 * grp + 1]; }
    float mx[8], sm[8];
    for (int v = 0; v < 8; v++) { mx[v] = -3.0e38f; sm[v] = 0.0f; }
    for (int m0 = 0; m0 < N0_; m0 += 16) {
        v8f acc = vzero8();
        const float* sb = sh + (long)(m0 + ln) * AVD_;
        for (int kk = 0; kk < 32; kk++) {
            v2f b; b.x = sb[kk * 4 + 2 * grp]; b.y = sb[kk * 4 + 2 * grp + 1];
            acc = wmma4(afr[kk], b, acc);
        }
        for (int v = 0; v < 8; v++) {
            int n = n0 + v + 8 * grp, m = m0 + ln;
            float x = eluf(acc[v]);
            float val = (adj[(long)(N0_ + n) * N_TOT + m] > 0.0f) ? x : NEG_BIG;
            float tmax = val;
            for (int d = 8; d >= 1; d >>= 1) tmax = fmaxf(tmax, __shfl_xor(tmax, d, 32));
            float p = expf(val - tmax);
            for (int d = 8; d >= 1; d >>= 1) p += __shfl_xor(p, d, 32);
            float nm = fmaxf(mx[v], tmax);
            sm[v] = sm[v] * expf(mx[v] - nm) + p * expf(tmax - nm);
            mx[v] = nm;
        }
    }
    if (ln == 0) {
        for (int v = 0; v < 8; v++) {
            int n = n0 + v + 8 * grp;
            rowmax[h * N0_ + n] = mx[v];
            rowsum[h * N0_ + n] = sm[v];
        }
    }
}

// ---------------- attention call 2: pass B (att @ t0 via LDS-staged tiles) ----------------
__global__ void k_att2_out(const float* __restrict__ g, const float* __restrict__ hAll,
                           const float* __restrict__ adj, const float* __restrict__ rowmax,
                           const float* __restrict__ rowsum, const float* __restrict__ t0,
                           float* __restrict__ outp) {
    __shared__ float att[16 * 132];          // padded stride 132 -> conflict-free
    int n0 = blockIdx.x * 16;
    int tid = threadIdx.x, wave = tid >> 5, l = tid & 31;
    int ln = l & 15, grp = l >> 4;
    int f0 = wave * 32;
    v8f acc0 = vzero8(), acc1 = vzero8();
    for (int h = 0; h < 4; h++) {
        const float* gh = g + (long)h * N_TOT * AVD_;
        const float* sh = hAll + (long)h * N_TOT * AVD_;
        float rmx[8], rinv[8];
        for (int v = 0; v < 8; v++) {
            int n = n0 + v + 8 * grp;
            rmx[v] = rowmax[h * N0_ + n];
            rinv[v] = 1.0f / rowsum[h * N0_ + n];
        }
        v2f afr[32];
        const float* ga = gh + (long)(N0_ + n0 + ln) * AVD_;
        for (int kk = 0; kk < 32; kk++) { afr[kk].x = ga[kk * 4 + 2 * grp]; afr[kk].y = ga[kk * 4 + 2 * grp + 1]; }
        for (int m0 = 0; m0 < N0_; m0 += 128) {
            int mt = m0 + wave * 16;
            v8f e = vzero8();
            const float* sb = sh + (long)(mt + ln) * AVD_;
            for (int kk = 0; kk < 32; kk++) {
                v2f b; b.x = sb[kk * 4 + 2 * grp]; b.y = sb[kk * 4 + 2 * grp + 1];
                e = wmma4(afr[kk], b, e);
            }
            for (int v = 0; v < 8; v++) {
                int n = n0 + v + 8 * grp, m = mt + ln;
                float x = eluf(e[v]);
                float val = (adj[(long)(N0_ + n) * N_TOT + m] > 0.0f) ? x : NEG_BIG;
                att[(v + 8 * grp) * 132 + wave * 16 + ln] = expf(val - rmx[v]) * rinv[v];
            }
            __syncthreads();
            // acc(16 x 32 f-chunk) += att(16x128) @ t0[m0..m0+127][f0..f0+31]
            for (int kk = 0; kk < 128; kk += 4) {
                int k0 = kk + 2 * grp;
                v2f a2; a2.x = att[ln * 132 + k0]; a2.y = att[ln * 132 + k0 + 1];
                const float* tr0 = t0 + (long)(m0 + k0) * HID_;
                const float* tr1 = t0 + (long)(m0 + k0 + 1) * HID_;
                v2f b0; b0.x = tr0[f0 + ln];      b0.y = tr1[f0 + ln];
                v2f b1; b1.x = tr0[f0 + 16 + ln]; b1.y = tr1[f0 + 16 + ln];
                acc0 = wmma4(a2, b0, acc0);
                acc1 = wmma4(a2, b1, acc1);
            }
            __syncthreads();
        }
    }
    for (int v = 0; v < 8; v++) {
        int n = n0 + v + 8 * grp;
        outp[(long)(N0_ + n) * HID_ + f0 + ln] = acc0[v] * 0.25f;
        outp[(long)(N0_ + n) * HID_ + f0 + 16 + ln] = acc1[v] * 0.25f;
    }
}

// ---------------- finalize scalars ----------------
__global__ void k_finalize(const float* __restrict__ scal, float* __restrict__ d_out) {
    // scal = {S00, S11, S01, loss_acc}
    d_out[(long)N_TOT * HID_ + 0] = scal[3] / (float)(NDROP_ * HID_);
    d_out[(long)N_TOT * HID_ + 1] =
        0.002f * (2.0f * logf(scal[2]) - logf(scal[0]) - logf(scal[1]));
}

// ---------------- host launch ----------------
extern "C" void kernel_launch(void* const* d_in, const int* in_sizes, int n_in,
                              void* d_out, int out_size, void* d_ws, size_t ws_size,
                              hipStream_t stream) {
    (void)in_sizes; (void)n_in; (void)out_size; (void)ws_size;
    const float* adj     = (const float*)d_in[0];
    const float* feat0   = (const float*)d_in[1];
    const float* feat1   = (const float*)d_in[2];
    const float* emb     = (const float*)d_in[3];
    const int*   keep    = (const int*)d_in[4];
    const int*   drop    = (const int*)d_in[5];
    const float* enc1_W  = (const float*)d_in[6];
    const float* enc1_b  = (const float*)d_in[7];
    const float* enc2a_W = (const float*)d_in[8];
    const float* enc2a_b = (const float*)d_in[9];
    const float* enc2b_W = (const float*)d_in[10];
    const float* enc2b_b = (const float*)d_in[11];
    const float* ac_W    = (const float*)d_in[12];   // (4,64,128)
    const float* ac_W2   = (const float*)d_in[13];   // (4,128,128)
    float* out = (float*)d_out;

    float* w = (float*)d_ws;
    size_t off = 0;
    float* t0     = w + off; off += (size_t)N0_ * HID_;          // 1,048,576
    float* t1     = w + off; off += (size_t)N0_ * HID_;
    float* featA  = w + off; off += (size_t)N1_ * HID_;
    float* featB  = w + off; off += (size_t)N1_ * (D1_ - HID_);
    float* hAll   = w + off; off += (size_t)4 * N_TOT * AVD_;
    float* gbuf   = w + off; off += (size_t)4 * N_TOT * AVD_;
    float* featK  = w + off; off += (size_t)NKEEP_ * HID_;
    float* rowmax = w + off; off += (size_t)4 * N0_;
    float* rowsum = w + off; off += (size_t)4 * N0_;
    float* nrm0   = w + off; off += N0_;
    float* nrm1   = w + off; off += N0_;
    float* stds   = w + off; off += D1_;
    float* scal   = w + off; off += 4;                           // {S00,S11,S01,loss}
    int*   colpos = (int*)(w + off); off += D1_;
    int*   istop  = (int*)(w + off); off += D1_;

    hipMemsetAsync(scal, 0, 4 * sizeof(float), stream);

    // --- feature selection for feat1 ---
    k_colstats<<<D1_, 256, 0, stream>>>(feat1, stds);
    k_rank<<<4, 256, 0, stream>>>(stds, colpos, istop);
    k_delpos<<<4, 256, 0, stream>>>(istop, colpos);
    k_gather<<<(N1_ * D1_) / 256, 256, 0, stream>>>(feat1, istop, colpos, featA, featB);

    // --- encoders (WMMA GEMM) ---
    k_gemm16<<<dim3(HID_ / 16, N0_ / 16, 1), 32, 0, stream>>>(
        feat0, enc1_W, enc1_b, t0, N0_, HID_, D0_, 1, 0, 0, 0);
    k_gemm16<<<dim3(HID_ / 16, N1_ / 16, 1), 32, 0, stream>>>(
        featA, enc2a_W, enc2a_b, t1, N1_, HID_, HID_, 0, 0, 0, 0);
    k_gemm16<<<dim3(HID_ / 16, N1_ / 16, 1), 32, 0, stream>>>(
        featB, enc2b_W, enc2b_b, t1, N1_, HID_, D1_ - HID_, 2, 0, 0, 0);

    // --- contrast loss (3 exp-sim sums; S(p1,p0)==S(p0,p1)) ---
    k_rownorm<<<N0_, 32, 0, stream>>>(t0, nrm0);
    k_rownorm<<<N1_, 32, 0, stream>>>(t1, nrm1);
    dim3 simg(N0_ / 16, N0_ / 16, 1);
    k_simsum<<<simg, 32, 0, stream>>>(t0, t0, nrm0, nrm0, &scal[0]);
    k_simsum<<<simg, 32, 0, stream>>>(t1, t1, nrm1, nrm1, &scal[1]);
    k_simsum<<<simg, 32, 0, stream>>>(t0, t1, nrm0, nrm1, &scal[2]);

    // --- attention projections: hAll[h] = emb @ W[h]; g[h] = hAll[h] @ W2[h] ---
    k_gemm16<<<dim3(AVD_ / 16, N_TOT / 16, 4), 32, 0, stream>>>(
        emb, ac_W, nullptr, hAll, N_TOT, AVD_, EMB_, 0,
        0, (long)EMB_ * AVD_, (long)N_TOT * AVD_);
    k_gemm16<<<dim3(AVD_ / 16, N_TOT / 16, 4), 32, 0, stream>>>(
        hAll, ac_W2, nullptr, gbuf, N_TOT, AVD_, AVD_, 0,
        (long)N_TOT * AVD_, (long)AVD_ * AVD_, (long)N_TOT * AVD_);

    // --- call 1: recon at drop rows + loss_ac ---
    k_gatherK<<<(NKEEP_ * HID_) / 256, 256, 0, stream>>>(t0, keep, featK);
    k_recon_loss<<<NDROP_, 256, 0, stream>>>(gbuf, hAll, adj, keep, drop, featK, t0,
                                             &scal[3]);

    // --- call 2: streaming masked softmax attention (WMMA both passes) ---
    k_att2_stats<<<dim3(N0_ / 16, 4), 32, 0, stream>>>(gbuf, hAll, adj, rowmax, rowsum);
    k_att2_out<<<N0_ / 16, 256, 0, stream>>>(gbuf, hAll, adj, rowmax, rowsum, t0, out);

    // --- assemble outputs ---
    hipMemcpyAsync(out, t0, (size_t)N0_ * HID_ * sizeof(float),
                   hipMemcpyDeviceToDevice, stream);
    k_finalize<<<1, 1, 0, stream>>>(scal, out);
}